// TransformerEncoder_7413113553432
// MI455X (gfx1250) — compile-verified
//
#include <hip/hip_runtime.h>
#include <cmath>

// ---------------------------------------------------------------------------
// Sizes (fixed by the reference)
// ---------------------------------------------------------------------------
#define N_LAYERS 4
#define BB 2
#define SS 2048
#define DD 1024
#define HH 16
#define HDIM 64
#define FDIM 4096
#define MM (BB * SS)   /* 4096 rows of activations */
#define D3 (3 * DD)    /* 3072 */
#define PAD 20         /* LDS row pitch in floats: bank-conflict-free AND 16B aligned */

typedef float v2f __attribute__((ext_vector_type(2)));
typedef float v8f __attribute__((ext_vector_type(8)));
typedef int v4i __attribute__((vector_size(16)));

// fp32 WMMA: D(16x16 f32) = A(16x4 f32) * B(4x16 f32) + C
static __device__ __forceinline__ v8f wmma_f32_16x16x4(v2f a, v2f b, v8f c) {
  return __builtin_amdgcn_wmma_f32_16x16x4_f32(false, a, false, b, (short)0, c,
                                               false, false);
}

// ---------------------------------------------------------------------------
// gfx1250 async global->LDS staging (ASYNCcnt-tracked); falls back to
// load+ds_store if the builtin is unavailable on this toolchain.
// ---------------------------------------------------------------------------
#if defined(__has_builtin)
#if __has_builtin(__builtin_amdgcn_global_load_async_to_lds_b128)
#define HAVE_ASYNC_LDS 1
#endif
#endif
#ifndef HAVE_ASYNC_LDS
#define HAVE_ASYNC_LDS 0
#endif

static __device__ __forceinline__ void stage_b128(const float* gsrc,
                                                  float* lds_dst) {
#if HAVE_ASYNC_LDS
  __builtin_amdgcn_global_load_async_to_lds_b128(
      (__attribute__((address_space(1))) v4i*)(void*)gsrc,
      (__attribute__((address_space(3))) v4i*)(void*)lds_dst, 0, 0);
#else
  *reinterpret_cast<float4*>(lds_dst) =
      *reinterpret_cast<const float4*>(gsrc);
#endif
}

static __device__ __forceinline__ void wait_stage() {
#if HAVE_ASYNC_LDS
#if __has_builtin(__builtin_amdgcn_s_wait_asynccnt)
  __builtin_amdgcn_s_wait_asynccnt(0);
#else
  asm volatile("s_wait_asynccnt 0x0" ::: "memory");
#endif
#endif
}

// ---------------------------------------------------------------------------
// RMSNorm: y = x / (||x|| * D^-0.5 + eps) * scale + offset   (one block / row)
// ---------------------------------------------------------------------------
__global__ __launch_bounds__(256) void rmsnorm_kernel(
    const float* __restrict__ x, const float* __restrict__ s,
    const float* __restrict__ o, float* __restrict__ y) {
  const int row = blockIdx.x;
  const int tid = threadIdx.x;
  const float* xr = x + (size_t)row * DD;
  float v[4];
  float ss = 0.0f;
#pragma unroll
  for (int i = 0; i < 4; ++i) {
    v[i] = xr[tid + i * 256];
    ss += v[i] * v[i];
  }
  __shared__ float red[256];
  red[tid] = ss;
  __syncthreads();
  for (int off = 128; off > 0; off >>= 1) {
    if (tid < off) red[tid] += red[tid + off];
    __syncthreads();
  }
  const float norm = sqrtf(red[0]);
  const float inv = 1.0f / (norm * 0.03125f + 1e-8f);  // D^-0.5 = 1/32
  float* yr = y + (size_t)row * DD;
#pragma unroll
  for (int i = 0; i < 4; ++i) {
    const int d = tid + i * 256;
    yr[d] = v[i] * inv * s[d] + o[d];
  }
}

// ---------------------------------------------------------------------------
// RoPE in place on qkv[B,S,3D], faithful to the reference's cos(emb)/sin(emb)
// of the precomputed sin/cos table.
// ---------------------------------------------------------------------------
__global__ __launch_bounds__(256) void rope_kernel(float* __restrict__ qkv) {
  const int idx = blockIdx.x * 256 + threadIdx.x;  // B*S*H*32 = 2^21 exact
  const int j = idx & 31;
  const int h = (idx >> 5) & 15;
  const int s = (idx >> 9) & 2047;
  const int b = idx >> 20;
  const float invf = exp2f(-(float)j * (13.287712379549449f / 32.0f));
  const float si = (float)s * invf;
  const float elo = sinf(si);
  const float ehi = cosf(si);
  const float clo = cosf(elo), slo = sinf(elo);
  const float chi = cosf(ehi), shi = sinf(ehi);
  const size_t base = ((size_t)(b * SS + s)) * D3 + h * (3 * HDIM);
  float qlo = qkv[base + j], qhi = qkv[base + 32 + j];
  qkv[base + j]      = qlo * clo - qhi * slo;
  qkv[base + 32 + j] = qhi * chi + qlo * shi;
  float klo = qkv[base + 64 + j], khi = qkv[base + 96 + j];
  qkv[base + 64 + j] = klo * clo - khi * slo;
  qkv[base + 96 + j] = khi * chi + klo * shi;
}

// ---------------------------------------------------------------------------
// WMMA GEMM:  C[M,N] = A[M,K] * W[N,K]^T + bias[N]  (+ residual | exact gelu)
// Block tile 128x128, BK=16 staged in padded LDS (async-to-LDS when present),
// 8 waves in a 4x2 grid, each wave owns a 32x64 register tile (2x4 WMMA accs)
// -> 8 v_wmma per 6 ds_load_b64 per k-step.
// mode: 0 = bias, 1 = bias+residual, 2 = bias+gelu
// ---------------------------------------------------------------------------
__global__ __launch_bounds__(256) void gemm_wmma_kernel(
    const float* __restrict__ A, const float* __restrict__ W,
    const float* __restrict__ bias, const float* __restrict__ res,
    float* __restrict__ C, int M, int N, int K, int mode) {
  __shared__ float As[128 * PAD];
  __shared__ float Bs[128 * PAD];
  const int tid = threadIdx.x;
  const int lane = tid & 31;
  const int w = tid >> 5;
  const int wm = w & 3;       // 4 M-strips of 32
  const int wn = w >> 2;      // 2 N-strips of 64
  const int m0 = blockIdx.y * 128;
  const int n0 = blockIdx.x * 128;
  const int lm = lane & 15;
  const int hi = lane >> 4;
  const int kb = hi * 2;

  // cooperative tile-load mapping: 128x16 floats = 512 float4 slots, 2/thread
  const int srow0 = tid >> 2;            // rows 0..63
  const int srow1 = srow0 + 64;          // rows 64..127
  const int scb = (tid & 3) * 4;         // col block (floats), 16B aligned

  v8f acc[2][4] = {};

  for (int kc = 0; kc < K; kc += 16) {
    stage_b128(A + (size_t)(m0 + srow0) * K + kc + scb, &As[srow0 * PAD + scb]);
    stage_b128(A + (size_t)(m0 + srow1) * K + kc + scb, &As[srow1 * PAD + scb]);
    stage_b128(W + (size_t)(n0 + srow0) * K + kc + scb, &Bs[srow0 * PAD + scb]);
    stage_b128(W + (size_t)(n0 + srow1) * K + kc + scb, &Bs[srow1 * PAD + scb]);
    wait_stage();
    __syncthreads();

#pragma unroll
    for (int kk = 0; kk < 16; kk += 4) {
      v2f af[2], bf[4];
#pragma unroll
      for (int i = 0; i < 2; ++i) {
        const int m = wm * 32 + i * 16 + lm;
        af[i] = *reinterpret_cast<const v2f*>(&As[m * PAD + kk + kb]);
      }
#pragma unroll
      for (int j = 0; j < 4; ++j) {
        const int n = wn * 64 + j * 16 + lm;
        bf[j] = *reinterpret_cast<const v2f*>(&Bs[n * PAD + kk + kb]);
      }
#pragma unroll
      for (int i = 0; i < 2; ++i)
#pragma unroll
        for (int j = 0; j < 4; ++j)
          acc[i][j] = wmma_f32_16x16x4(af[i], bf[j], acc[i][j]);
    }
    __syncthreads();
  }

  // epilogue
#pragma unroll
  for (int i = 0; i < 2; ++i) {
#pragma unroll
    for (int j = 0; j < 4; ++j) {
      const int gcol = n0 + wn * 64 + j * 16 + lm;
      const float bv = bias[gcol];
#pragma unroll
      for (int r = 0; r < 8; ++r) {
        const int grow = m0 + wm * 32 + i * 16 + r + 8 * hi;
        float v = acc[i][j][r] + bv;
        if (mode == 1) {
          v += res[(size_t)grow * N + gcol];
        } else if (mode == 2) {
          v = 0.5f * v * (1.0f + erff(v * 0.7071067811865476f));
        }
        C[(size_t)grow * N + gcol] = v;
      }
    }
  }
}

// ---------------------------------------------------------------------------
// Flash-style attention. One wave per (b, h, 16 query rows). qkv layout is
// [B,S,3D] with q at h*192+0, k at +64, v at +128. Output vals[B,S,D] with
// column h*64+d (heads re-merged for the Wo GEMM).
// ---------------------------------------------------------------------------
__global__ __launch_bounds__(32) void attn_kernel(
    const float* __restrict__ qkv, float* __restrict__ vals) {
  const int lane = threadIdx.x;
  const int qt = blockIdx.x & 127;       // S/16 = 128 query tiles
  const int bh = blockIdx.x >> 7;
  const int b = bh >> 4;
  const int h = bh & 15;
  const int s0 = qt * 16;
  const int lm = lane & 15;
  const int hi = lane >> 4;
  const int kb = hi * 2;

  // Q fragment (16x64), pre-scaled by 1/sqrt(HD) = 0.125
  v2f qa[16];
  {
    const size_t qbase = ((size_t)(b * SS + s0 + lm)) * D3 + h * (3 * HDIM);
#pragma unroll
    for (int t = 0; t < 16; ++t) {
      const float* p = qkv + qbase + t * 4 + kb;
      qa[t].x = p[0] * 0.125f;
      qa[t].y = p[1] * 0.125f;
    }
  }

  v8f o[4] = {};
  float mst[8], lst[8];
#pragma unroll
  for (int r = 0; r < 8; ++r) { mst[r] = -1e30f; lst[r] = 0.0f; }

  __shared__ float ps[16 * PAD];

  for (int n0 = 0; n0 < SS; n0 += 16) {
    // logits tile S = Q * K^T (already scaled via Q)
    v8f c = {};
    {
      const size_t kbase =
          ((size_t)(b * SS + n0 + lm)) * D3 + h * (3 * HDIM) + HDIM;
#pragma unroll
      for (int t = 0; t < 16; ++t) {
        const float* p = qkv + kbase + t * 4 + kb;
        v2f kf;
        kf.x = p[0];
        kf.y = p[1];
        c = wmma_f32_16x16x4(qa[t], kf, c);
      }
    }
    // online softmax update; each C VGPR holds one 16-wide row per lane-half
#pragma unroll
    for (int r = 0; r < 8; ++r) {
      const float cv = c[r];
      float rm = cv;
      rm = fmaxf(rm, __shfl_xor(rm, 1));
      rm = fmaxf(rm, __shfl_xor(rm, 2));
      rm = fmaxf(rm, __shfl_xor(rm, 4));
      rm = fmaxf(rm, __shfl_xor(rm, 8));
      const float mnew = fmaxf(mst[r], rm);
      const float alpha = __expf(mst[r] - mnew);
      const float pv = __expf(cv - mnew);
      float rs = pv;
      rs += __shfl_xor(rs, 1);
      rs += __shfl_xor(rs, 2);
      rs += __shfl_xor(rs, 4);
      rs += __shfl_xor(rs, 8);
      lst[r] = lst[r] * alpha + rs;
      mst[r] = mnew;
#pragma unroll
      for (int j = 0; j < 4; ++j) o[j][r] *= alpha;
      ps[(r + 8 * hi) * PAD + lm] = pv;  // C-layout -> LDS
    }
    __syncthreads();
    // re-read P in A-layout (8B-aligned thanks to PAD=20)
    v2f pa[4];
#pragma unroll
    for (int t = 0; t < 4; ++t) {
      pa[t] = *reinterpret_cast<const v2f*>(&ps[lm * PAD + t * 4 + kb]);
    }
    // O += P * V  (V tile is 16 keys x 64 dims)
    {
      const size_t vrow = ((size_t)(b * SS + n0)) * D3 + h * (3 * HDIM) + 2 * HDIM;
#pragma unroll
      for (int j = 0; j < 4; ++j) {
#pragma unroll
        for (int t = 0; t < 4; ++t) {
          const size_t a0 = vrow + (size_t)(t * 4 + kb) * D3 + j * 16 + lm;
          v2f vb;
          vb.x = qkv[a0];
          vb.y = qkv[a0 + D3];
          o[j] = wmma_f32_16x16x4(pa[t], vb, o[j]);
        }
      }
    }
    __syncthreads();  // protect ps before next tile overwrites it
  }

  // normalize and write out: vals[b, s0+row, h*64 + d]
#pragma unroll
  for (int r = 0; r < 8; ++r) {
    const float invl = 1.0f / lst[r];
    const int row = s0 + r + 8 * hi;
    const size_t base = ((size_t)(b * SS + row)) * DD + h * HDIM;
#pragma unroll
    for (int j = 0; j < 4; ++j) {
      vals[base + j * 16 + lm] = o[j][r] * invl;
    }
  }
}

// ---------------------------------------------------------------------------
// Host-side orchestration
// ---------------------------------------------------------------------------
extern "C" void kernel_launch(void* const* d_in, const int* in_sizes, int n_in,
                              void* d_out, int out_size, void* d_ws,
                              size_t ws_size, hipStream_t stream) {
  (void)in_sizes; (void)n_in; (void)out_size; (void)ws_size;
  const float* x    = (const float*)d_in[0];
  const float* s1   = (const float*)d_in[1];
  const float* o1   = (const float*)d_in[2];
  const float* Wqkv = (const float*)d_in[3];
  const float* bqkv = (const float*)d_in[4];
  const float* Wo   = (const float*)d_in[5];
  const float* bo   = (const float*)d_in[6];
  const float* s2   = (const float*)d_in[7];
  const float* o2   = (const float*)d_in[8];
  const float* W1   = (const float*)d_in[9];
  const float* b1   = (const float*)d_in[10];
  const float* W2   = (const float*)d_in[11];
  const float* b2   = (const float*)d_in[12];

  float* ws = (float*)d_ws;
  const size_t actN = (size_t)MM * DD;       // 4,194,304 floats
  float* X    = ws;                           // layer output (ping)
  float* H1   = X + actN;                     // post-norm1
  float* HRES = H1 + actN;                    // attn residual sum
  float* H2   = HRES + actN;                  // post-norm2
  float* QKV  = H2 + actN;                    // [M, 3D]
  float* VALS = QKV + (size_t)MM * D3;        // [M, D]
  float* FFB  = QKV;  // FFN buffer [M, F] overlaps dead QKV+VALS (exact fit)

  const float* xin = x;
  for (int l = 0; l < N_LAYERS; ++l) {
    rmsnorm_kernel<<<MM, 256, 0, stream>>>(xin, s1 + l * DD, o1 + l * DD, H1);
    gemm_wmma_kernel<<<dim3(D3 / 128, MM / 128), 256, 0, stream>>>(
        H1, Wqkv + (size_t)l * D3 * DD, bqkv + (size_t)l * D3, nullptr, QKV,
        MM, D3, DD, 0);
    rope_kernel<<<(BB * SS * HH * 32) / 256, 256, 0, stream>>>(QKV);
    attn_kernel<<<BB * HH * (SS / 16), 32, 0, stream>>>(QKV, VALS);
    gemm_wmma_kernel<<<dim3(DD / 128, MM / 128), 256, 0, stream>>>(
        VALS, Wo + (size_t)l * DD * DD, bo + (size_t)l * DD, H1, HRES,
        MM, DD, DD, 1);
    rmsnorm_kernel<<<MM, 256, 0, stream>>>(HRES, s2 + l * DD, o2 + l * DD, H2);
    gemm_wmma_kernel<<<dim3(FDIM / 128, MM / 128), 256, 0, stream>>>(
        H2, W1 + (size_t)l * FDIM * DD, b1 + (size_t)l * FDIM, nullptr, FFB,
        MM, FDIM, DD, 2);
    float* outp = (l == N_LAYERS - 1) ? (float*)d_out : X;
    gemm_wmma_kernel<<<dim3(DD / 128, MM / 128), 256, 0, stream>>>(
        FFB, W2 + (size_t)l * DD * FDIM, b2 + (size_t)l * DD, H2, outp,
        MM, DD, FDIM, 1);
    xin = X;
  }
}